// LightGCN_40424232190055
// MI455X (gfx1250) — compile-verified
//
#include <hip/hip_runtime.h>

// LightGCN on MI455X (gfx1250, wave32).
// N = U + I nodes, D = 64, E edges, 3 propagation layers.
// Pipeline: deg count -> rsqrt -> edge norms -> 3x atomic scatter layers
// -> fused mean-of-4-layers via V_WMMA_F32_16X16X4_F32 (A = all 0.25).

typedef __attribute__((ext_vector_type(2))) float v2f;
typedef __attribute__((ext_vector_type(8))) float v8f;

#define D_EMB 64

// ---------------------------------------------------------------- zero
__global__ void zero_f32(float* __restrict__ p, long n) {
  long i = (long)blockIdx.x * blockDim.x + threadIdx.x;
  long stride = (long)gridDim.x * blockDim.x;
  for (; i < n; i += stride) p[i] = 0.0f;
}

// ---------------------------------------------------------------- degree
__global__ void count_deg(const long long* __restrict__ col,
                          float* __restrict__ deg, int E) {
  int i = blockIdx.x * blockDim.x + threadIdx.x;
  int stride = gridDim.x * blockDim.x;
  for (; i < E; i += stride) {
    atomicAdd(&deg[(long)col[i]], 1.0f);
  }
}

__global__ void deg_inv_sqrt_k(const float* __restrict__ deg,
                               float* __restrict__ dinv, int n) {
  int i = blockIdx.x * blockDim.x + threadIdx.x;
  int stride = gridDim.x * blockDim.x;
  for (; i < n; i += stride) {
    float d = deg[i];
    dinv[i] = (d > 0.0f) ? rsqrtf(d) : 0.0f;   // d >= 1 when > 0
  }
}

__global__ void norm_k(const long long* __restrict__ row,
                       const long long* __restrict__ col,
                       const float* __restrict__ dinv,
                       float* __restrict__ nrm, int E) {
  int i = blockIdx.x * blockDim.x + threadIdx.x;
  int stride = gridDim.x * blockDim.x;
  for (; i < E; i += stride) {
    nrm[i] = dinv[(long)row[i]] * dinv[(long)col[i]];
  }
}

// ---------------------------------------------------------------- scatter
// One wave32 per edge; each lane handles 2 consecutive dims (float2 gather,
// two global_atomic_add_f32 scatters). Layer 1 gathers from the two input
// arrays directly (implicit concat), later layers from a ws buffer.
__global__ void scatter_first(const float* __restrict__ user,
                              const float* __restrict__ item, long U,
                              const long long* __restrict__ row,
                              const long long* __restrict__ col,
                              const float* __restrict__ nrm,
                              float* __restrict__ dst, int E) {
  int lane = threadIdx.x & 31;
  long wave = ((long)blockIdx.x * blockDim.x + threadIdx.x) >> 5;
  long nwaves = ((long)gridDim.x * blockDim.x) >> 5;
  for (long e = wave; e < E; e += nwaves) {
    long r = (long)row[e];
    long c = (long)col[e];
    float w = nrm[e];
    const float* src = (r < U) ? (user + r * D_EMB)
                               : (item + (r - U) * D_EMB);
    float2 v = *(const float2*)(src + lane * 2);
    float* d = dst + c * D_EMB + lane * 2;
    atomicAdd(d,     v.x * w);
    atomicAdd(d + 1, v.y * w);
  }
}

__global__ void scatter_mid(const float* __restrict__ xsrc,
                            const long long* __restrict__ row,
                            const long long* __restrict__ col,
                            const float* __restrict__ nrm,
                            float* __restrict__ dst, int E) {
  int lane = threadIdx.x & 31;
  long wave = ((long)blockIdx.x * blockDim.x + threadIdx.x) >> 5;
  long nwaves = ((long)gridDim.x * blockDim.x) >> 5;
  for (long e = wave; e < E; e += nwaves) {
    long r = (long)row[e];
    long c = (long)col[e];
    float w = nrm[e];
    float2 v = *(const float2*)(xsrc + r * D_EMB + lane * 2);
    float* d = dst + c * D_EMB + lane * 2;
    atomicAdd(d,     v.x * w);
    atomicAdd(d + 1, v.y * w);
  }
}

// ---------------------------------------------------------------- mean (WMMA)
// out[j] = 0.25*(x0 + x1 + x2 + x3)[j] as a K=4 f32 WMMA contraction.
// A (16x4) = all 0.25: lanes 0-15 hold K0/K1 in a.x/a.y, lanes 16-31 K2/K3.
// B (4x16): VGPR0 lanes0-15 = row K=0 (x0), lanes16-31 = K=2 (x2);
//           VGPR1 lanes0-15 = K=1 (x1), lanes16-31 = K=3 (x3).
// D row 0 (VGPR0, lanes 0-15) carries the 16 means for this tile.
__global__ void mean_wmma(const float* __restrict__ user,
                          const float* __restrict__ item, long Uel,
                          const float* __restrict__ x1,
                          const float* __restrict__ x2,
                          const float* __restrict__ x3,
                          float* __restrict__ out, long nTiles) {
  int lane = threadIdx.x & 31;
  long wave = ((long)blockIdx.x * blockDim.x + threadIdx.x) >> 5;
  long nwaves = ((long)gridDim.x * blockDim.x) >> 5;
  v2f a;
  a.x = 0.25f;
  a.y = 0.25f;
  for (long t = wave; t < nTiles; t += nwaves) {
    long j = t * 16 + (lane & 15);
    v2f b;
    if (lane < 16) {
      b.x = (j < Uel) ? user[j] : item[j - Uel];  // x0 = concat(user, item)
      b.y = x1[j];
    } else {
      b.x = x2[j];
      b.y = x3[j];
    }
    v8f c = {};
    v8f d = __builtin_amdgcn_wmma_f32_16x16x4_f32(
        /*neg_a=*/false, a, /*neg_b=*/false, b,
        /*c_mod=*/(short)0, c, /*reuse_a=*/false, /*reuse_b=*/false);
    if (lane < 16) out[j] = d[0];
  }
}

// ---------------------------------------------------------------- launch
static inline dim3 grid_for(long work, int per_block) {
  long g = (work + per_block - 1) / per_block;
  if (g < 1) g = 1;
  if (g > 65536) g = 65536;
  return dim3((unsigned)g);
}

extern "C" void kernel_launch(void* const* d_in, const int* in_sizes, int n_in,
                              void* d_out, int out_size, void* d_ws, size_t ws_size,
                              hipStream_t stream) {
  const float* user = (const float*)d_in[0];
  const float* item = (const float*)d_in[1];
  const long long* ei = (const long long*)d_in[2];

  long Uel = in_sizes[0];          // U * 64
  long U = Uel / D_EMB;
  long I = (long)in_sizes[1] / D_EMB;
  long N = U + I;
  int E = in_sizes[2] / 2;
  long ND = N * D_EMB;

  const long long* row = ei;        // edge_index[0]
  const long long* col = ei + E;    // edge_index[1]

  float* ws = (float*)d_ws;
  float* xA = ws;                   // layer-1 output  [N*64]
  float* xB = ws + ND;              // layer-2 output  [N*64]
  float* xC = ws + 2 * ND;          // layer-3 output  [N*64]
  float* deg = ws + 3 * ND;         // [N]
  float* dinv = deg + N;            // [N]
  float* nrm = dinv + N;            // [E]

  dim3 blk(256);

  // zero xA, xB, xC, deg (contiguous)
  zero_f32<<<grid_for(3 * ND + N, 256 * 8), blk, 0, stream>>>(ws, 3 * ND + N);

  count_deg<<<grid_for(E, 256), blk, 0, stream>>>(col, deg, E);
  deg_inv_sqrt_k<<<grid_for(N, 256), blk, 0, stream>>>(deg, dinv, (int)N);
  norm_k<<<grid_for(E, 256), blk, 0, stream>>>(row, col, dinv, nrm, E);

  // 3 propagation layers (one wave per edge)
  scatter_first<<<grid_for((long)E * 32, 256), blk, 0, stream>>>(
      user, item, U, row, col, nrm, xA, E);
  scatter_mid<<<grid_for((long)E * 32, 256), blk, 0, stream>>>(
      xA, row, col, nrm, xB, E);
  scatter_mid<<<grid_for((long)E * 32, 256), blk, 0, stream>>>(
      xB, row, col, nrm, xC, E);

  // fused mean of {x0, x1, x2, x3} via v_wmma_f32_16x16x4_f32
  long nTiles = ND / 16;
  mean_wmma<<<grid_for(nTiles * 32, 256), blk, 0, stream>>>(
      user, item, Uel, xA, xB, xC, (float*)d_out, nTiles);
}